// SG4_3496103379564
// MI455X (gfx1250) — compile-verified
//
#include <hip/hip_runtime.h>
#include <hip/hip_bf16.h>

#define NN   100000
#define NE   3200000
#define INC  256
#define HIDC 32
#define OUTC 64

typedef __attribute__((ext_vector_type(2))) float v2f;
typedef __attribute__((ext_vector_type(8))) float v8f;

// ---------------- degree / normalization ----------------

__global__ void deg_init_k(float* __restrict__ deg) {
    int i = blockIdx.x * blockDim.x + threadIdx.x;
    if (i < NN) deg[i] = 1.0f;            // self-loop contributes 1
}

__global__ void deg_acc_k(const long long* __restrict__ col, float* __restrict__ deg) {
    int e = blockIdx.x * blockDim.x + threadIdx.x;
    if (e < NE) {
        int c = (int)col[e];
        __hip_atomic_fetch_add(&deg[c], 1.0f, __ATOMIC_RELAXED, __HIP_MEMORY_SCOPE_AGENT);
    }
}

__global__ void dinv_k(const float* __restrict__ deg, float* __restrict__ dinv) {
    int i = blockIdx.x * blockDim.x + threadIdx.x;
    if (i < NN) dinv[i] = rsqrtf(deg[i]); // deg >= 1 always
}

// ---------------- propagation (gather-scale-scatter) ----------------

// self-loop term: p[i][:] = dinv[i]^2 * h[i][:]
__global__ void prop_init_k(const float* __restrict__ h, const float* __restrict__ dinv,
                            float* __restrict__ p) {
    int t = blockIdx.x * blockDim.x + threadIdx.x;   // t < NN*32 exactly covered by grid
    int i = t >> 5;
    float d = dinv[i];
    p[t] = d * d * h[t];
}

// one (edge, channel) pair per lane: a wave handles one edge's 32 channels
__global__ void prop_edge_k(const long long* __restrict__ row,
                            const long long* __restrict__ col,
                            const float* __restrict__ dinv,
                            const float* __restrict__ h,
                            float* __restrict__ p) {
    int t = blockIdx.x * blockDim.x + threadIdx.x;   // grid sized to NE*32 exactly
    int e = t >> 5;
    int c = t & 31;
    int r  = (int)row[e];
    int cl = (int)col[e];
    float w = dinv[r] * dinv[cl];
    float v = w * h[(size_t)r * HIDC + c];
    __hip_atomic_fetch_add(&p[(size_t)cl * HIDC + c], v,
                           __ATOMIC_RELAXED, __HIP_MEMORY_SCOPE_AGENT);
}

// ---------------- fp32 WMMA GEMM: out[M,NOUT] = A[M,KIN] @ W[NOUT,KIN]^T + b ----------------
// One wave per 16-row tile. NOUT/16 accumulators. KIN multiple of 4, M multiple of 16.

template <int KIN, int NOUT, bool RELU>
__global__ __launch_bounds__(256) void gemm_wmma_k(const float* __restrict__ A,
                                                   const float* __restrict__ W,
                                                   const float* __restrict__ bias,
                                                   float* __restrict__ out) {
    constexpr int TILES = NN / 16;                       // 6250
    int wave = blockIdx.x * (blockDim.x >> 5) + (threadIdx.x >> 5);
    if (wave >= TILES) return;                           // whole wave exits -> EXEC all-1 for WMMA

    const int lane = threadIdx.x & 31;
    const int lr   = lane & 15;       // row/col within tile
    const int hi   = lane >> 4;       // lane half
    const int kh   = hi << 1;         // K offset 0 or 2 (32-bit A/B layout)
    const int m0   = wave << 4;

    constexpr int NT = NOUT / 16;
    v8f acc[NT];
    #pragma unroll
    for (int nt = 0; nt < NT; ++nt) {
        float bv = bias[nt * 16 + lr];                   // C[M][N] = b[N]
        #pragma unroll
        for (int v = 0; v < 8; ++v) acc[nt][v] = bv;
    }

    for (int k = 0; k < KIN; k += 4) {
        // A fragment: lane holds A[m0+lr][k+kh], A[m0+lr][k+kh+1] (contiguous pair)
        const float* ap = A + (size_t)(m0 + lr) * KIN + k + kh;
        v2f a; a.x = ap[0]; a.y = ap[1];
        #pragma unroll
        for (int nt = 0; nt < NT; ++nt) {
            // B fragment: B[k'][n] = W[n][k'] ; lane holds W[nt*16+lr][k+kh .. +1]
            const float* bp = W + (size_t)(nt * 16 + lr) * KIN + k + kh;
            v2f b; b.x = bp[0]; b.y = bp[1];
            acc[nt] = __builtin_amdgcn_wmma_f32_16x16x4_f32(
                false, a, false, b, (short)0, acc[nt], false, false);
        }
    }

    // C/D layout: VGPR v, low lanes -> M = v, high lanes -> M = v+8 ; N = lr (+16*nt)
    #pragma unroll
    for (int nt = 0; nt < NT; ++nt) {
        #pragma unroll
        for (int v = 0; v < 8; ++v) {
            float val = acc[nt][v];
            if (RELU) val = fmaxf(val, 0.0f);
            out[(size_t)(m0 + hi * 8 + v) * NOUT + nt * 16 + lr] = val;
        }
    }
}

// ---------------- launch ----------------

extern "C" void kernel_launch(void* const* d_in, const int* in_sizes, int n_in,
                              void* d_out, int out_size, void* d_ws, size_t ws_size,
                              hipStream_t stream) {
    const float*     x    = (const float*)d_in[0];
    const long long* ei   = (const long long*)d_in[1];   // int64 edge_index, shape (2, NE) flat
    const float*     W0   = (const float*)d_in[2];
    const float*     b0   = (const float*)d_in[3];
    const float*     W1   = (const float*)d_in[4];
    const float*     b1   = (const float*)d_in[5];
    const float*     W2   = (const float*)d_in[6];
    const float*     b2   = (const float*)d_in[7];
    const float*     W3   = (const float*)d_in[8];
    const float*     b3   = (const float*)d_in[9];
    const float*     W4   = (const float*)d_in[10];
    const float*     b4   = (const float*)d_in[11];
    float*           out  = (float*)d_out;

    const long long* row = ei;            // edge_index[0]
    const long long* col = ei + NE;       // edge_index[1]

    // workspace carve-out (floats): hA | hB | deg | dinv
    float* hA   = (float*)d_ws;
    float* hB   = hA + (size_t)NN * HIDC;
    float* deg  = hB + (size_t)NN * HIDC;
    float* dinv = deg + NN;

    const int BLK = 256;
    const int gNode   = (NN + BLK - 1) / BLK;        // 391
    const int gEdge   = NE / BLK;                    // 12500 (exact)
    const int gFeat   = (NN * HIDC) / BLK;           // 12500 (exact)
    const int gEdgeCh = (NE * HIDC) / BLK;           // 400000 (exact)
    const int gGemm   = ((NN / 16) + 7) / 8;         // 782 blocks of 8 waves

    // GCN normalization
    deg_init_k<<<gNode, BLK, 0, stream>>>(deg);
    deg_acc_k <<<gEdge, BLK, 0, stream>>>(col, deg);
    dinv_k    <<<gNode, BLK, 0, stream>>>(deg, dinv);

    // h = x @ W0^T + b0
    gemm_wmma_k<INC, HIDC, false><<<gGemm, BLK, 0, stream>>>(x, W0, b0, hA);

    // three SGConv layers: prop then linear+relu
    const float* Ws[3] = {W1, W2, W3};
    const float* bs[3] = {b1, b2, b3};
    for (int l = 0; l < 3; ++l) {
        prop_init_k<<<gFeat,   BLK, 0, stream>>>(hA, dinv, hB);
        prop_edge_k<<<gEdgeCh, BLK, 0, stream>>>(row, col, dinv, hA, hB);
        gemm_wmma_k<HIDC, HIDC, true><<<gGemm, BLK, 0, stream>>>(hB, Ws[l], bs[l], hA);
    }

    // out = h @ W4^T + b4
    gemm_wmma_k<HIDC, OUTC, false><<<gGemm, BLK, 0, stream>>>(hA, W4, b4, out);
}